// BeamSearch_70961449664776
// MI455X (gfx1250) — compile-verified
//
#include <hip/hip_runtime.h>
#include <hip/hip_bf16.h>

// ---------------- problem constants (from reference) ----------------
#define HID 512
#define EIN 256
#define SLEN 50
#define DD 1024         // D
#define VV 50000        // vocab
#define KB 3            // beam width
#define LL 20           // max length
#define BOS_T 1
#define EOS_T 2
#define PAD_T 0

// ---------------- workspace layout (float indices) ----------------
#define OFF_H    0                      // 3*1024   decoder h
#define OFF_C    3072                   // 3*1024   decoder c
#define OFF_Z    6144                   // 3*4096   lstm gates
#define OFF_H2   18432                  // 3*1024
#define OFF_C2   21504                  // 3*1024
#define OFF_ZC   24576                  // 3*1024   tanh(cat @ Wcat.T)
#define OFF_PROB 27648                  // 3
#define OFF_PART 27680                  // NBLK_V*3*8 partial softmax/topk records
#define INT_BASE 40960                  // int region starts here (32-bit words)
// int region: last[3] @0, fin[3] @3, tokens[3*20] @8

#define NTILE_V 3125                    // 50000/16
#define NBLK_V  391                     // ceil(3125/8)

// logits kernel TDM tiling
#define AROWS   128                     // rows per block tile
#define ACOLS   64                      // cols per chunk
#define NCHUNK  16                      // 1024/64
#define ASTRIDE 68                      // LDS row stride: 64 data + 4 pad dwords

typedef __attribute__((ext_vector_type(2))) float v2f;
typedef __attribute__((ext_vector_type(8))) float v8f;
typedef __attribute__((ext_vector_type(4))) unsigned int u32x4;
typedef __attribute__((ext_vector_type(4))) int i32x4;
typedef __attribute__((ext_vector_type(8))) int i32x8;

__device__ __forceinline__ float sigm(float x) { return 1.f / (1.f + __expf(-x)); }

__device__ __forceinline__ unsigned lds_off(const void* p) {
    // flat LDS address = {aperture[63:32], offset[31:0]} -> low 32 bits are the LDS byte offset
    return (unsigned)(uintptr_t)p;
}

// TDM: load a 2D tile (tile_rows x ACOLS f32, row stride = rs elems) from
// global into LDS with 4-dword padding after every 64 dwords (row stride 68).
// rows_valid = tensor_dim1 from tile start: rows past it are zero-filled.
__device__ __forceinline__ void tdm_load_tile(unsigned lds_byte, const float* gsrc,
                                              unsigned rows_valid, unsigned rs)
{
    unsigned long long ga = (unsigned long long)(uintptr_t)gsrc;
    u32x4 g0;
    g0[0] = 1u;                                                 // count=1 (valid), user mode
    g0[1] = lds_byte;                                           // lds_addr
    g0[2] = (unsigned)(ga & 0xffffffffu);                       // global_addr[31:0]
    g0[3] = (unsigned)((ga >> 32) & 0x01ffffffu) | (2u << 30);  // global_addr[56:32] | type=2
    i32x8 g1;
    g1[0] = (int)((2u << 16)        // data_size = 4 bytes
                | (1u << 20)        // pad_enable
                | (5u << 22)        // pad_interval: 64 dwords
                | (3u << 25));      // pad_amount: 4 dwords
    g1[1] = (int)((ACOLS & 0xffffu) << 16);                     // tensor_dim0 lo16 (=64)
    g1[2] = (int)((rows_valid & 0xffffu) << 16);                // dim0 hi16=0 | tensor_dim1 lo16
    g1[3] = (int)(((rows_valid >> 16) & 0xffffu) | (ACOLS << 16)); // tensor_dim1 hi16 | tile_dim0
    g1[4] = (int)(AROWS & 0xffffu);                             // tile_dim1 | tile_dim2=0
    g1[5] = (int)rs;                                            // tensor_dim0_stride lo32
    g1[6] = 0;                                                  // stride0 hi16 | stride1 lo16
    g1[7] = 0;                                                  // stride1 hi32
    i32x4 z4 = {};
#if defined(__clang_major__) && __clang_major__ >= 23
    i32x8 z8 = {};
    __builtin_amdgcn_tensor_load_to_lds(g0, g1, z4, z4, z8, 0);
#else
    __builtin_amdgcn_tensor_load_to_lds(g0, g1, z4, z4, 0);
#endif
}

// =====================================================================
// Encoder: 50 sequential forward LSTM steps + 1 backward step (B=1).
// hb[-1] of the reference == backward cell applied once to sentence[49].
// =====================================================================
__global__ __launch_bounds__(512) void enc_kernel(
    const float* __restrict__ x,
    const float* __restrict__ Wih_f, const float* __restrict__ Whh_f, const float* __restrict__ b_f,
    const float* __restrict__ Wih_b, const float* __restrict__ Whh_b, const float* __restrict__ b_b,
    float* __restrict__ wf, int* __restrict__ wi)
{
    __shared__ float sh[HID];
    __shared__ float sc[HID];
    __shared__ float sx[EIN];
    const int tid = threadIdx.x;

    if (tid < HID) { sh[tid] = 0.f; sc[tid] = 0.f; }
    __syncthreads();

    for (int t = 0; t < SLEN; ++t) {
        if (tid < EIN) sx[tid] = x[t * EIN + tid];
        __syncthreads();
        float hn = 0.f, cn = 0.f;
        if (tid < HID) {
            float zi = b_f[tid], zf = b_f[HID + tid], zg = b_f[2 * HID + tid], zo = b_f[3 * HID + tid];
            const float* ri = Wih_f + (size_t)tid * EIN;
            const float* rf = Wih_f + (size_t)(HID + tid) * EIN;
            const float* rg = Wih_f + (size_t)(2 * HID + tid) * EIN;
            const float* ro = Wih_f + (size_t)(3 * HID + tid) * EIN;
            #pragma unroll 4
            for (int k = 0; k < EIN; ++k) {
                float xv = sx[k];
                zi = fmaf(ri[k], xv, zi); zf = fmaf(rf[k], xv, zf);
                zg = fmaf(rg[k], xv, zg); zo = fmaf(ro[k], xv, zo);
            }
            const float* qi = Whh_f + (size_t)tid * HID;
            const float* qf = Whh_f + (size_t)(HID + tid) * HID;
            const float* qg = Whh_f + (size_t)(2 * HID + tid) * HID;
            const float* qo = Whh_f + (size_t)(3 * HID + tid) * HID;
            #pragma unroll 4
            for (int k = 0; k < HID; ++k) {
                float hv = sh[k];
                zi = fmaf(qi[k], hv, zi); zf = fmaf(qf[k], hv, zf);
                zg = fmaf(qg[k], hv, zg); zo = fmaf(qo[k], hv, zo);
            }
            cn = sigm(zf) * sc[tid] + sigm(zi) * tanhf(zg);
            hn = sigm(zo) * tanhf(cn);
        }
        __syncthreads();
        if (tid < HID) { sh[tid] = hn; sc[tid] = cn; }
        __syncthreads();
    }

    if (tid < EIN) sx[tid] = x[(SLEN - 1) * EIN + tid];
    __syncthreads();
    float hb = 0.f;
    if (tid < HID) {
        float zi = b_b[tid], zg = b_b[2 * HID + tid], zo = b_b[3 * HID + tid];
        const float* ri = Wih_b + (size_t)tid * EIN;
        const float* rg = Wih_b + (size_t)(2 * HID + tid) * EIN;
        const float* ro = Wih_b + (size_t)(3 * HID + tid) * EIN;
        #pragma unroll 4
        for (int k = 0; k < EIN; ++k) {
            float xv = sx[k];
            zi = fmaf(ri[k], xv, zi); zg = fmaf(rg[k], xv, zg); zo = fmaf(ro[k], xv, zo);
        }
        float cb = sigm(zi) * tanhf(zg);       // c=0: f-gate term vanishes
        hb = sigm(zo) * tanhf(cb);
    }

    if (tid < HID) {
        for (int n = 0; n < KB; ++n) {
            wf[OFF_H + n * DD + tid]       = sh[tid];
            wf[OFF_H + n * DD + HID + tid] = hb;
            wf[OFF_C + n * DD + tid]       = 0.f;
            wf[OFF_C + n * DD + HID + tid] = 0.f;
        }
    }
    if (tid == 0) {
        wf[OFF_PROB + 0] = 1.f; wf[OFF_PROB + 1] = 0.f; wf[OFF_PROB + 2] = 0.f;
        for (int n = 0; n < KB; ++n) { wi[n] = BOS_T; wi[3 + n] = 0; }
        for (int n = 0; n < KB; ++n) {
            wi[8 + n * LL] = BOS_T;
            for (int j = 1; j < LL; ++j) wi[8 + n * LL + j] = PAD_T;
        }
    }
}

// =====================================================================
// Decoder LSTM gates: z[n][m] = dot([e;h], [dWih[m];dWhh[m]]) + db[m]
// WMMA f32 16x16x4; xs padded with a zero row so B-fragment LDS loads
// are unconditional (no EXEC-mask churn in the inner loop).
// =====================================================================
__global__ __launch_bounds__(256) void gates_kernel(
    const float* __restrict__ dWih, const float* __restrict__ dWhh,
    const float* __restrict__ db, const float* __restrict__ emb,
    float* __restrict__ wf, const int* __restrict__ wi)
{
    __shared__ float xs[KB + 1][2 * DD + 4];
    const int tid = threadIdx.x;
    for (int i = tid; i < KB * 2 * DD; i += 256) {
        int n = i / (2 * DD), k = i % (2 * DD);
        xs[n][k] = (k < DD) ? emb[(size_t)wi[n] * DD + k]
                            : wf[OFF_H + n * DD + (k - DD)];
    }
    for (int i = tid; i < 2 * DD; i += 256) xs[KB][i] = 0.f;
    __syncthreads();

    const int lane = tid & 31, wid = tid >> 5;
    const int tile = blockIdx.x * 8 + wid;      // 0..255
    const int m0   = tile * 16;
    const int mrow = m0 + (lane & 15);
    const int klo  = (lane >> 4) * 2;
    const int nb   = lane & 15;
    const int nbc  = (nb < KB) ? nb : KB;       // lanes >= KB read the zero row

    v8f acc = {};
    #pragma unroll 4
    for (int k0 = 0; k0 < DD; k0 += 4) {
        v2f a = *(const v2f*)(dWih + (size_t)mrow * DD + k0 + klo);
        v2f b = *(const v2f*)(&xs[nbc][k0 + klo]);
        acc = __builtin_amdgcn_wmma_f32_16x16x4_f32(false, a, false, b, (short)0, acc, false, false);
    }
    #pragma unroll 4
    for (int k0 = 0; k0 < DD; k0 += 4) {
        v2f a = *(const v2f*)(dWhh + (size_t)mrow * DD + k0 + klo);
        v2f b = *(const v2f*)(&xs[nbc][DD + k0 + klo]);
        acc = __builtin_amdgcn_wmma_f32_16x16x4_f32(false, a, false, b, (short)0, acc, false, false);
    }

    const int nb2  = (lane < 16) ? lane : (lane - 16);
    const int mofs = (lane < 16) ? 0 : 8;
    if (nb2 < KB) {
        #pragma unroll
        for (int r = 0; r < 8; ++r) {
            int m = m0 + mofs + r;
            wf[OFF_Z + nb2 * 4 * DD + m] = acc[r] + db[m];
        }
    }
}

// =====================================================================
// Gate combine -> h2, c2
// =====================================================================
__global__ void combine_kernel(float* __restrict__ wf)
{
    int i = blockIdx.x * blockDim.x + threadIdx.x;
    if (i >= KB * DD) return;
    int n = i / DD, j = i % DD;
    const float* z = wf + OFF_Z + (size_t)n * 4 * DD;
    float zi = z[j], zf = z[DD + j], zg = z[2 * DD + j], zo = z[3 * DD + j];
    float c  = wf[OFF_C + i];
    float c2 = sigm(zf) * c + sigm(zi) * tanhf(zg);
    wf[OFF_C2 + i] = c2;
    wf[OFF_H2 + i] = sigm(zo) * tanhf(c2);
}

// =====================================================================
// zc = tanh(Wcat @ [h2; e] + bcat), 64 M-tiles -> 8 blocks x 8 waves
// =====================================================================
__global__ __launch_bounds__(256) void zcat_kernel(
    const float* __restrict__ Wcat, const float* __restrict__ bcat,
    const float* __restrict__ emb, float* __restrict__ wf, const int* __restrict__ wi)
{
    __shared__ float xs[KB + 1][2 * DD + 4];
    const int tid = threadIdx.x;
    for (int i = tid; i < KB * 2 * DD; i += 256) {
        int n = i / (2 * DD), k = i % (2 * DD);
        xs[n][k] = (k < DD) ? wf[OFF_H2 + n * DD + k]
                            : emb[(size_t)wi[n] * DD + (k - DD)];
    }
    for (int i = tid; i < 2 * DD; i += 256) xs[KB][i] = 0.f;
    __syncthreads();

    const int lane = tid & 31, wid = tid >> 5;
    const int tile = blockIdx.x * 8 + wid;      // 0..63
    const int m0   = tile * 16;
    const int mrow = m0 + (lane & 15);
    const int klo  = (lane >> 4) * 2;
    const int nb   = lane & 15;
    const int nbc  = (nb < KB) ? nb : KB;

    v8f acc = {};
    #pragma unroll 4
    for (int k0 = 0; k0 < 2 * DD; k0 += 4) {
        v2f a = *(const v2f*)(Wcat + (size_t)mrow * 2 * DD + k0 + klo);
        v2f b = *(const v2f*)(&xs[nbc][k0 + klo]);
        acc = __builtin_amdgcn_wmma_f32_16x16x4_f32(false, a, false, b, (short)0, acc, false, false);
    }

    const int nb2  = (lane < 16) ? lane : (lane - 16);
    const int mofs = (lane < 16) ? 0 : 8;
    if (nb2 < KB) {
        #pragma unroll
        for (int r = 0; r < 8; ++r) {
            int m = m0 + mofs + r;
            wf[OFF_ZC + nb2 * DD + m] = tanhf(acc[r] + bcat[m]);
        }
    }
}

// =====================================================================
// Big pass: logits = Wout @ zc + bout.  TDM double-buffers 128x64 f32
// tiles of Wout into LDS (padded rows -> conflict-free ds_load_b64),
// WMMA consumes, fused online-softmax partials + per-block top-3.
// =====================================================================
__global__ __launch_bounds__(256) void logits_kernel(
    const float* __restrict__ Wout, const float* __restrict__ bout,
    float* __restrict__ wf)
{
    __shared__ float xs[KB + 1][DD + 4];
    __shared__ float Abuf[2][AROWS * ASTRIDE];   // 2 x 34816 B
    __shared__ float ll[AROWS][4];
    const int tid = threadIdx.x;
    for (int i = tid; i < KB * DD; i += 256) xs[i / DD][i % DD] = wf[OFF_ZC + i];
    for (int i = tid; i < DD; i += 256) xs[KB][i] = 0.f;

    const int lane = tid & 31, wid = tid >> 5;
    const int row0 = blockIdx.x * AROWS;                // global tile row
    int rows_here = VV - row0; if (rows_here > AROWS) rows_here = AROWS;

    if (wid == 0)
        tdm_load_tile(lds_off(&Abuf[0][0]), Wout + (size_t)row0 * DD, (unsigned)rows_here, DD);

    const int klo = (lane >> 4) * 2;
    const int nb  = lane & 15;
    const int nbc = (nb < KB) ? nb : KB;
    const int lrow = wid * 16 + (lane & 15);

    v8f acc = {};
    for (int c = 0; c < NCHUNK; ++c) {
        if (wid == 0) {
            if (c + 1 < NCHUNK) {
                tdm_load_tile(lds_off(&Abuf[(c + 1) & 1][0]),
                              Wout + (size_t)row0 * DD + (c + 1) * ACOLS,
                              (unsigned)rows_here, DD);
                __builtin_amdgcn_s_wait_tensorcnt(1);   // chunk c complete (in-order)
            } else {
                __builtin_amdgcn_s_wait_tensorcnt(0);
            }
        }
        __syncthreads();                                // tile c visible to all waves
        const float* A = &Abuf[c & 1][0];
        const int cb = c * ACOLS;
        #pragma unroll 4
        for (int k0 = 0; k0 < ACOLS; k0 += 4) {
            v2f a = *(const v2f*)(A + lrow * ASTRIDE + k0 + klo);
            v2f b = *(const v2f*)(&xs[nbc][cb + k0 + klo]);
            acc = __builtin_amdgcn_wmma_f32_16x16x4_f32(false, a, false, b, (short)0, acc, false, false);
        }
        __syncthreads();                                // done with tile c before overwrite
    }

    const int nb2  = (lane < 16) ? lane : (lane - 16);
    const int mofs = (lane < 16) ? 0 : 8;
    if (nb2 < KB) {
        #pragma unroll
        for (int r = 0; r < 8; ++r) {
            int lr = wid * 16 + mofs + r;
            int m  = row0 + lr;
            float bv = (m < VV) ? bout[m] : 0.f;
            ll[lr][nb2] = acc[r] + bv;
        }
    }
    __syncthreads();

    if (tid < KB) {
        const int nrows = rows_here;
        const int base  = row0;
        float bm = -3.4e38f;
        for (int r = 0; r < nrows; ++r) { float v = ll[r][tid]; if (v > bm) bm = v; }
        float bs = 0.f;
        float tv0 = -3.4e38f, tv1 = -3.4e38f, tv2 = -3.4e38f;
        int   ti0 = 0, ti1 = 0, ti2 = 0;
        for (int r = 0; r < nrows; ++r) {
            float v = ll[r][tid];
            bs += __expf(v - bm);
            if (v > tv0)      { tv2 = tv1; ti2 = ti1; tv1 = tv0; ti1 = ti0; tv0 = v; ti0 = base + r; }
            else if (v > tv1) { tv2 = tv1; ti2 = ti1; tv1 = v;  ti1 = base + r; }
            else if (v > tv2) { tv2 = v;  ti2 = base + r; }
        }
        float* p = wf + OFF_PART + ((size_t)blockIdx.x * KB + tid) * 8;
        p[0] = bm; p[1] = bs; p[2] = tv0; p[3] = tv1; p[4] = tv2;
        ((int*)p)[5] = ti0; ((int*)p)[6] = ti1; ((int*)p)[7] = ti2;
    }
}

// =====================================================================
// Finalize one beam step: merge partials -> softmax top-3 per beam,
// combine/prune to K of 9 (stable ties like jax top_k), permute state.
// =====================================================================
__global__ __launch_bounds__(128) void finalize_kernel(
    float* __restrict__ wf, int* __restrict__ wi,
    float* __restrict__ out, int t)
{
    __shared__ float cv[KB][KB];
    __shared__ int   cidx[KB][KB];
    __shared__ int   s_par[KB], s_tok[KB], s_fin[KB];
    __shared__ float s_tp[KB];
    __shared__ int   told[KB * LL];
    const int tid = threadIdx.x;

    if (tid < KB) {
        const float* base = wf + OFF_PART;
        float gm = -3.4e38f;
        for (int b = 0; b < NBLK_V; ++b) {
            float v = base[((size_t)b * KB + tid) * 8];
            if (v > gm) gm = v;
        }
        float gs = 0.f;
        float tv0 = -3.4e38f, tv1 = -3.4e38f, tv2 = -3.4e38f;
        int   ti0 = 0, ti1 = 0, ti2 = 0;
        for (int b = 0; b < NBLK_V; ++b) {
            const float* p = base + ((size_t)b * KB + tid) * 8;
            gs += p[1] * __expf(p[0] - gm);
            for (int j = 0; j < 3; ++j) {
                float v = p[2 + j]; int ix = ((const int*)p)[5 + j];
                if (v > tv0)      { tv2 = tv1; ti2 = ti1; tv1 = tv0; ti1 = ti0; tv0 = v; ti0 = ix; }
                else if (v > tv1) { tv2 = tv1; ti2 = ti1; tv1 = v;  ti1 = ix; }
                else if (v > tv2) { tv2 = v;  ti2 = ix; }
            }
        }
        float inv = 1.f / gs;
        cv[tid][0] = __expf(tv0 - gm) * inv; cidx[tid][0] = ti0;
        cv[tid][1] = __expf(tv1 - gm) * inv; cidx[tid][1] = ti1;
        cv[tid][2] = __expf(tv2 - gm) * inv; cidx[tid][2] = ti2;
    }
    if (tid == 0) {
        for (int i = 0; i < KB * LL; ++i) told[i] = wi[8 + i];
    }
    __syncthreads();

    if (tid == 0) {
        float cp[KB * KB]; int ci[KB * KB];
        for (int n = 0; n < KB; ++n) {
            int f = wi[3 + n];
            float pr = wf[OFF_PROB + n];
            for (int j = 0; j < KB; ++j) {
                cp[n * KB + j] = f ? (j == 0 ? pr : 0.f) : pr * cv[n][j];
                ci[n * KB + j] = f ? PAD_T : cidx[n][j];
            }
        }
        bool used[KB * KB];
        for (int q = 0; q < KB * KB; ++q) used[q] = false;
        for (int s = 0; s < KB; ++s) {
            int best = 0; float bv = -3.4e38f;
            for (int q = 0; q < KB * KB; ++q)
                if (!used[q] && cp[q] > bv) { bv = cp[q]; best = q; }
            used[best] = true;
            int pa = best / KB, tk = ci[best];
            int fp = wi[3 + pa];
            s_par[s] = pa; s_tok[s] = tk; s_tp[s] = bv;
            s_fin[s] = (fp || tk == EOS_T || (tk == PAD_T && fp)) ? 1 : 0;
        }
    }
    __syncthreads();

    for (int i = tid; i < KB * DD; i += 128) {
        int n = i / DD, d = i % DD;
        int pa = s_par[n];
        wf[OFF_H + i] = wf[OFF_H2 + pa * DD + d];
        wf[OFF_C + i] = wf[OFF_C2 + pa * DD + d];
    }
    if (tid < KB * LL) {
        int n = tid / LL, j = tid % LL;
        int v = told[s_par[n] * LL + j];
        if (j == t) v = s_tok[n];
        wi[8 + n * LL + j] = v;
        out[n * LL + j] = (float)v;
    }
    if (tid < KB) {
        wf[OFF_PROB + tid] = s_tp[tid];
        wi[tid]     = s_tok[tid];   // last
        wi[3 + tid] = s_fin[tid];   // fin
        out[KB * LL + tid] = s_tp[tid];
    }
}

// =====================================================================
extern "C" void kernel_launch(void* const* d_in, const int* in_sizes, int n_in,
                              void* d_out, int out_size, void* d_ws, size_t ws_size,
                              hipStream_t stream)
{
    (void)in_sizes; (void)n_in; (void)out_size; (void)ws_size;
    const float* sentence = (const float*)d_in[0];
    const float* eWih_f   = (const float*)d_in[1];
    const float* eWhh_f   = (const float*)d_in[2];
    const float* eb_f     = (const float*)d_in[3];
    const float* eWih_b   = (const float*)d_in[4];
    const float* eWhh_b   = (const float*)d_in[5];
    const float* eb_b     = (const float*)d_in[6];
    const float* emb      = (const float*)d_in[7];
    const float* dWih     = (const float*)d_in[8];
    const float* dWhh     = (const float*)d_in[9];
    const float* db       = (const float*)d_in[10];
    const float* Wcat     = (const float*)d_in[11];
    const float* bcat     = (const float*)d_in[12];
    const float* Wout     = (const float*)d_in[13];
    const float* bout     = (const float*)d_in[14];

    float* F = (float*)d_ws;
    int*   I = (int*)d_ws + INT_BASE;
    float* out = (float*)d_out;

    enc_kernel<<<1, 512, 0, stream>>>(sentence, eWih_f, eWhh_f, eb_f,
                                      eWih_b, eWhh_b, eb_b, F, I);

    for (int t = 1; t < LL; ++t) {
        gates_kernel<<<32, 256, 0, stream>>>(dWih, dWhh, db, emb, F, I);
        combine_kernel<<<(KB * DD + 255) / 256, 256, 0, stream>>>(F);
        zcat_kernel<<<8, 256, 0, stream>>>(Wcat, bcat, emb, F, I);
        logits_kernel<<<NBLK_V, 256, 0, stream>>>(Wout, bout, F);
        finalize_kernel<<<1, 128, 0, stream>>>(F, I, out, t);
    }
}